// SAGE1_63651415326799
// MI455X (gfx1250) — compile-verified
//
#include <hip/hip_runtime.h>

#define NN 100000
#define NE 1600000
#define HID 128
#define NCLS 40

typedef float v2f __attribute__((ext_vector_type(2)));
typedef float v8f __attribute__((ext_vector_type(8)));

// ---------------- degree kernels ----------------
__global__ void deg_count_kernel(const int* __restrict__ dst, float* __restrict__ deg, int n_edges) {
    int e = blockIdx.x * blockDim.x + threadIdx.x;
    if (e < n_edges) unsafeAtomicAdd(&deg[dst[e]], 1.0f);
}

__global__ void deg_invert_kernel(float* __restrict__ deg, int n) {
    int i = blockIdx.x * blockDim.x + threadIdx.x;
    if (i < n) deg[i] = 1.0f / fmaxf(deg[i], 1.0f);
}

// ---------------- edge scatter-add (mean numerator) ----------------
// 32 lanes per edge, each lane moves a float4 chunk of the 128-wide feature.
// agg (51.2MB) + X (51.2MB) are L2-resident on MI455X (192MB L2), so these
// atomics resolve at L2 bandwidth, not HBM.
__global__ void scatter_add_kernel(const float* __restrict__ X,
                                   const int* __restrict__ src,
                                   const int* __restrict__ dst,
                                   float* __restrict__ agg, int n_edges) {
    unsigned long long gid = (unsigned long long)blockIdx.x * blockDim.x + threadIdx.x;
    unsigned int e = (unsigned int)(gid >> 5);
    if (e >= (unsigned)n_edges) return;
    int c = ((int)gid & 31) << 2;                  // feature offset 0..124, step 4
    int s = src[e], d = dst[e];
    const float4 v = *(const float4*)(X + (size_t)s * HID + c);
    float* p = agg + (size_t)d * HID + c;
    unsafeAtomicAdd(p + 0, v.x);
    unsafeAtomicAdd(p + 1, v.y);
    unsafeAtomicAdd(p + 2, v.z);
    unsafeAtomicAdd(p + 3, v.w);
}

// ---------------- fused SAGE layer: OUT = X@Wself + (agg*invdeg)@Wneigh + b, epilogue ----------------
// grid.x = NN/16 row tiles; wave w of the block owns output columns [16w,16w+16).
// Uses V_WMMA_F32_16X16X4_F32 (exact fp32): per k-step of 4, two WMMAs (self + neigh).
// A layout (16x4 f32): lane half h holds k-pair {k+2h, k+2h+1} for row M=lane%16.
// B layout mirrors A with N=lane%16. C: VGPR r -> (M = r + 8*(lane/16), N = lane%16).
// X and OUT may alias (in-place layer): each tile reads its 16 rows fully before storing.
__global__ void sage_gemm_kernel(const float* X,
                                 const float* __restrict__ AGG,
                                 const float* __restrict__ invdeg,
                                 const float* __restrict__ Wself,
                                 const float* __restrict__ Wneigh,
                                 const float* __restrict__ bias,
                                 const unsigned char* __restrict__ mask,
                                 const int* __restrict__ g2,
                                 const float* __restrict__ gh,
                                 float* OUT,
                                 int outd, int do_epilogue) {
    const int wave = threadIdx.x >> 5;
    const int lane = threadIdx.x & 31;
    const int half = lane >> 4;          // 0: lanes 0-15, 1: lanes 16-31
    const int lrow = lane & 15;
    const int row0 = blockIdx.x << 4;
    const int row  = row0 + lrow;        // A-operand row (M = lane%16 for both halves)
    const int col  = (wave << 4) + lrow; // B-operand / C column
    const bool colok = col < outd;       // layer-3 pad guard (40 -> 48)
    const int   ccol = colok ? col : 0;  // clamped (always in-bounds loads)
    const float cmsk = colok ? 1.0f : 0.0f;

    const float id = invdeg[row];
    const float* xr = X   + (size_t)row * HID;
    const float* ar = AGG + (size_t)row * HID;

    v8f acc = {};
    #pragma unroll 4
    for (int k = 0; k < HID; k += 4) {
        const int ka = k + (half << 1);                 // k-pair base for this lane-half
        v2f a_s, a_n, b_s, b_n;
        a_s.x = xr[ka];          a_s.y = xr[ka + 1];
        a_n.x = ar[ka] * id;     a_n.y = ar[ka + 1] * id;
        const float* wsp = Wself  + (size_t)ka * outd + ccol;
        const float* wnp = Wneigh + (size_t)ka * outd + ccol;
        b_s.x = wsp[0]    * cmsk;
        b_s.y = wsp[outd] * cmsk;
        b_n.x = wnp[0]    * cmsk;
        b_n.y = wnp[outd] * cmsk;
        acc = __builtin_amdgcn_wmma_f32_16x16x4_f32(false, a_s, false, b_s, (short)0, acc, false, false);
        acc = __builtin_amdgcn_wmma_f32_16x16x4_f32(false, a_n, false, b_n, (short)0, acc, false, false);
    }

    if (colok) {
        const float bc = bias[col];
        #pragma unroll
        for (int r = 0; r < 8; ++r) {
            const int m = row0 + r + (half << 3);       // global output row
            float v = acc[r] + bc;
            if (do_epilogue) {
                if (!mask[m]) v = gh[(size_t)g2[m] * HID + col];
                v = fmaxf(v, 0.0f);
            }
            OUT[(size_t)m * outd + col] = v;
        }
    }
}

extern "C" void kernel_launch(void* const* d_in, const int* in_sizes, int n_in,
                              void* d_out, int out_size, void* d_ws, size_t ws_size,
                              hipStream_t stream) {
    const float* feat = (const float*)d_in[0];
    const float* gh1  = (const float*)d_in[1];
    const float* gh2  = (const float*)d_in[2];
    const float* W1s  = (const float*)d_in[3];
    const float* W1n  = (const float*)d_in[4];
    const float* b1   = (const float*)d_in[5];
    const float* W2s  = (const float*)d_in[6];
    const float* W2n  = (const float*)d_in[7];
    const float* b2   = (const float*)d_in[8];
    const float* W3s  = (const float*)d_in[9];
    const float* W3n  = (const float*)d_in[10];
    const float* b3   = (const float*)d_in[11];
    const int*   esrc = (const int*)d_in[12];
    const int*   edst = (const int*)d_in[13];
    const int*   g2   = (const int*)d_in[14];
    const unsigned char* mask = (const unsigned char*)d_in[15]; // jax bool = 1 byte
    float* out = (float*)d_out;

    char* ws = (char*)d_ws;
    const size_t FEAT_BYTES = (size_t)NN * HID * sizeof(float); // 51.2 MB
    float* agg = (float*)ws;                          // [NN,128]
    float* h   = (float*)(ws + FEAT_BYTES);           // [NN,128]
    float* idg = (float*)(ws + 2 * FEAT_BYTES);       // [NN]

    // degrees (same for all layers: dst-only)
    hipMemsetAsync(idg, 0, (size_t)NN * sizeof(float), stream);
    deg_count_kernel<<<(NE + 255) / 256, 256, 0, stream>>>(edst, idg, NE);
    deg_invert_kernel<<<(NN + 255) / 256, 256, 0, stream>>>(idg, NN);

    const int scat_blocks = (int)(((unsigned long long)NE * 32 + 255) / 256);
    const int gemm_blocks = NN / 16; // 6250, exact

    // ---- layer 1: feat -> h (mask-replace with gh1, relu) ----
    hipMemsetAsync(agg, 0, FEAT_BYTES, stream);
    scatter_add_kernel<<<scat_blocks, 256, 0, stream>>>(feat, esrc, edst, agg, NE);
    sage_gemm_kernel<<<gemm_blocks, 256, 0, stream>>>(feat, agg, idg, W1s, W1n, b1,
                                                      mask, g2, gh1, h, HID, 1);

    // ---- layer 2: h -> h in-place (mask-replace with gh2, relu) ----
    hipMemsetAsync(agg, 0, FEAT_BYTES, stream);
    scatter_add_kernel<<<scat_blocks, 256, 0, stream>>>(h, esrc, edst, agg, NE);
    sage_gemm_kernel<<<gemm_blocks, 256, 0, stream>>>(h, agg, idg, W2s, W2n, b2,
                                                      mask, g2, gh2, h, HID, 1);

    // ---- layer 3: h -> out (40 classes, 3 column-tile waves, no epilogue) ----
    hipMemsetAsync(agg, 0, FEAT_BYTES, stream);
    scatter_add_kernel<<<scat_blocks, 256, 0, stream>>>(h, esrc, edst, agg, NE);
    sage_gemm_kernel<<<gemm_blocks, 96, 0, stream>>>(h, agg, idg, W3s, W3n, b3,
                                                     mask, g2, gh1, out, NCLS, 0);
}